// Attention_1013612281902
// MI455X (gfx1250) — compile-verified
//
#include <hip/hip_runtime.h>

typedef __attribute__((ext_vector_type(2))) float v2f;
typedef __attribute__((ext_vector_type(8))) float v8f;

#define BM 128
#define BN 64
#define KT 32
#define KP (KT + 2)   // padded row stride (34 dwords): 8B-aligned rows, bank-conflict-free
#define PLANE_ELEMS (512 * 960)
#define NCH 64

// Generic batched f32 GEMM: D[z][m,n] = alpha * sum_k A[z][m,k] * B[z][k,n]
// Operand addressing fully stride-parameterized (transposes via strides).
// z = b*Hn + h ; per-operand batch strides for b and h.
// Requires M%BM==0, N%BN==0, K%KT==0 (guaranteed by host).
__global__ __launch_bounds__(256) void wmma_gemm_f32(
    const float* __restrict__ A, const float* __restrict__ Bmat,
    float* __restrict__ D,
    int M, int N, int K,
    long a_rs, long a_cs, long a_sb, long a_sh,
    long b_rs, long b_cs, long b_sb, long b_sh,
    long d_rs, long d_sb, long d_sh,
    int Hn, float alpha)
{
  // A tile [BM x KT] as As[m][k]; B tile stored TRANSPOSED as Bs[n][k] so the
  // per-fragment (k, k+1) pair is contiguous -> direct 64-bit LDS loads, no repack movs.
  __shared__ float As[BM][KP];
  __shared__ float Bs[BN][KP];

  const int tid = threadIdx.x;
  const int z = blockIdx.z;
  const int bb = z / Hn, hh = z % Hn;
  const float* Ap = A + a_sb * bb + a_sh * hh;
  const float* Bp = Bmat + b_sb * bb + b_sh * hh;
  float* Dp = D + d_sb * bb + d_sh * hh;

  const int m0 = blockIdx.y * BM;
  const int n0 = blockIdx.x * BN;

  const int lane = tid & 31;
  const int wave = tid >> 5;
  const int wm = (wave & 3) * 32;   // 4 waves along M (128)
  const int wn = (wave >> 2) * 32;  // 2 waves along N (64)
  const int lm = lane & 15;
  const int lk2 = (lane >> 4) << 1; // 0 for lanes 0-15, 2 for lanes 16-31

  const v8f vzero = {0.f, 0.f, 0.f, 0.f, 0.f, 0.f, 0.f, 0.f};
  v8f acc[2][2];
  acc[0][0] = vzero; acc[0][1] = vzero; acc[1][0] = vzero; acc[1][1] = vzero;

  for (int k0 = 0; k0 < K; k0 += KT) {
    __syncthreads();
    // ---- stage A tile [BM x KT] into LDS (coalescing-aware mapping) ----
    if (a_cs == 1) {
      for (int i = tid; i < BM * KT; i += 256) {
        int m = i >> 5, k = i & (KT - 1);
        As[m][k] = Ap[(long)(m0 + m) * a_rs + (long)(k0 + k)];
      }
    } else {
      for (int i = tid; i < BM * KT; i += 256) {
        int k = i >> 7, m = i & (BM - 1);
        As[m][k] = Ap[(long)(m0 + m) * a_rs + (long)(k0 + k) * a_cs];
      }
    }
    // ---- stage B tile into LDS, transposed: Bs[n][k] ----
    if (b_cs == 1) {
      for (int i = tid; i < KT * BN; i += 256) {
        int k = i >> 6, n = i & (BN - 1);
        Bs[n][k] = Bp[(long)(k0 + k) * b_rs + (long)(n0 + n)];
      }
    } else {
      for (int i = tid; i < KT * BN; i += 256) {
        int n = i >> 5, k = i & (KT - 1);
        Bs[n][k] = Bp[(long)(k0 + k) * b_rs + (long)(n0 + n) * b_cs];
      }
    }
    __syncthreads();

    // ---- 8 x (2x2) WMMA f32 16x16x4 steps over the K-tile ----
    #pragma unroll
    for (int kk = 0; kk < KT; kk += 4) {
      v2f af[2], bf[2];
      #pragma unroll
      for (int t = 0; t < 2; ++t) {
        // A 16x4 f32 layout: lanes 0-15 hold K=kk+0/kk+1; lanes 16-31 kk+2/kk+3.
        // Rows are 8B-aligned and kk+lk2 is even -> aligned 64-bit LDS loads.
        af[t] = *(const v2f*)&As[wm + t * 16 + lm][kk + lk2];
        // B 4x16 f32 layout mirrored (N across lanes); Bs is [n][k] so the
        // (k, k+1) pair is contiguous too.
        bf[t] = *(const v2f*)&Bs[wn + t * 16 + lm][kk + lk2];
      }
      #pragma unroll
      for (int tm = 0; tm < 2; ++tm)
        #pragma unroll
        for (int tn = 0; tn < 2; ++tn)
          acc[tm][tn] = __builtin_amdgcn_wmma_f32_16x16x4_f32(
              false, af[tm], false, bf[tn], (short)0, acc[tm][tn], false, false);
    }
  }

  // ---- store: 16x16 f32 C/D layout: VGPR g -> M=g (lanes 0-15) / M=8+g (16-31) ----
  const int mo = (lane < 16) ? 0 : 8;
  #pragma unroll
  for (int tm = 0; tm < 2; ++tm)
    #pragma unroll
    for (int tn = 0; tn < 2; ++tn) {
      const int gn = n0 + wn + tn * 16 + lm;
      #pragma unroll
      for (int g = 0; g < 8; ++g) {
        const int gm = m0 + wm + tm * 16 + mo + g;
        Dp[(long)gm * d_rs + gn] = alpha * acc[tm][tn][g];
      }
    }
}

// ---- InstanceNorm stats, stage 1: per-(plane, chunk) partial sum/sumsq ----
__global__ __launch_bounds__(256) void inorm_partial(const float* __restrict__ S,
                                                     float* __restrict__ part)
{
  __shared__ float r1[256], r2[256];
  const int plane = blockIdx.x;      // 0..15
  const int chunk = blockIdx.y;      // 0..NCH-1
  const int CH = PLANE_ELEMS / NCH;  // 7680
  const float* P = S + (long)plane * PLANE_ELEMS + (long)chunk * CH;
  float s = 0.f, s2 = 0.f;
  for (int i = threadIdx.x; i < CH; i += 256) {
    float v = P[i];
    s += v; s2 += v * v;
  }
  r1[threadIdx.x] = s; r2[threadIdx.x] = s2;
  __syncthreads();
  for (int o = 128; o > 0; o >>= 1) {
    if (threadIdx.x < o) {
      r1[threadIdx.x] += r1[threadIdx.x + o];
      r2[threadIdx.x] += r2[threadIdx.x + o];
    }
    __syncthreads();
  }
  if (threadIdx.x == 0) {
    part[(plane * NCH + chunk) * 2 + 0] = r1[0];
    part[(plane * NCH + chunk) * 2 + 1] = r2[0];
  }
}

// ---- InstanceNorm stats, stage 2: reduce NCH partials -> mean, rstd ----
__global__ __launch_bounds__(64) void inorm_finalize(const float* __restrict__ part,
                                                     float* __restrict__ stats)
{
  __shared__ float r1[64], r2[64];
  const int plane = blockIdx.x;
  r1[threadIdx.x] = part[(plane * NCH + threadIdx.x) * 2 + 0];
  r2[threadIdx.x] = part[(plane * NCH + threadIdx.x) * 2 + 1];
  __syncthreads();
  for (int o = 32; o > 0; o >>= 1) {
    if (threadIdx.x < o) {
      r1[threadIdx.x] += r1[threadIdx.x + o];
      r2[threadIdx.x] += r2[threadIdx.x + o];
    }
    __syncthreads();
  }
  if (threadIdx.x == 0) {
    const float inv = 1.0f / (float)PLANE_ELEMS;
    float mean = r1[0] * inv;
    float var = r2[0] * inv - mean * mean;
    stats[plane * 2 + 0] = mean;
    stats[plane * 2 + 1] = rsqrtf(fmaxf(var, 0.f) + 1e-5f);
  }
}

// ---- normalize + softmax over KV, one block per (b,h,c) row, in place ----
__global__ __launch_bounds__(256) void inorm_softmax(float* __restrict__ S,
                                                     const float* __restrict__ stats,
                                                     int Cdim, int KVdim)
{
  __shared__ float red[256];
  const long row = blockIdx.x;            // b*h*C + c
  const int plane = (int)(row / Cdim);
  const float mean = stats[plane * 2 + 0];
  const float rstd = stats[plane * 2 + 1];
  float* Rw = S + row * KVdim;

  float v[4];
  int cnt = 0;
  float mx = -3.0e38f;
  for (int i = threadIdx.x; i < KVdim; i += 256) {
    float x = (Rw[i] - mean) * rstd;
    v[cnt++] = x;
    mx = fmaxf(mx, x);
  }
  red[threadIdx.x] = mx; __syncthreads();
  for (int o = 128; o > 0; o >>= 1) {
    if (threadIdx.x < o) red[threadIdx.x] = fmaxf(red[threadIdx.x], red[threadIdx.x + o]);
    __syncthreads();
  }
  const float m = red[0];
  __syncthreads();
  float s = 0.f;
  for (int j = 0; j < cnt; ++j) { v[j] = __expf(v[j] - m); s += v[j]; }
  red[threadIdx.x] = s; __syncthreads();
  for (int o = 128; o > 0; o >>= 1) {
    if (threadIdx.x < o) red[threadIdx.x] += red[threadIdx.x + o];
    __syncthreads();
  }
  const float inv = 1.0f / red[0];
  cnt = 0;
  for (int i = threadIdx.x; i < KVdim; i += 256) Rw[i] = v[cnt++] * inv;
}

// ---- Pp[b] = sum_h SA[b,h]  (1/H already folded into GEMM alpha) ----
__global__ __launch_bounds__(256) void head_sum(const float* __restrict__ SA,
                                                float* __restrict__ Pp,
                                                long plane, int Hn)
{
  const long i = (long)blockIdx.x * 256 + threadIdx.x;  // over B*plane
  const long b = i / plane, r = i % plane;
  float s = 0.f;
  for (int h = 0; h < Hn; ++h) s += SA[(b * Hn + h) * plane + r];
  Pp[i] = s;
}

extern "C" void kernel_launch(void* const* d_in, const int* in_sizes, int n_in,
                              void* d_out, int out_size, void* d_ws, size_t ws_size,
                              hipStream_t stream)
{
  const int B = 4, S = 4096, C = 512, KV = 960, H = 4;
  const float* emb     = (const float*)d_in[0];
  const float* emb_all = (const float*)d_in[1];
  const float* Wq      = (const float*)d_in[2];
  const float* Wk      = (const float*)d_in[3];
  const float* Wv      = (const float*)d_in[4];
  const float* Wo      = (const float*)d_in[5];
  float* out = (float*)d_out;

  const long plane = (long)C * KV;  // 491520
  float* ws    = (float*)d_ws;
  float* G     = ws;                        // [B, C, KV]
  float* SA    = G  + (long)B * plane;      // [B, H, C, KV]
  float* SB    = SA + (long)B * H * plane;  // [B, H, C, KV]
  float* Pp    = SB + (long)B * H * plane;  // [B, C, KV]
  float* Rb    = Pp + (long)B * plane;      // [B, C, KV]
  float* stats = Rb + (long)B * plane;      // [B*H, 2]
  float* part  = stats + 2 * B * H;         // [B*H, NCH, 2]

  dim3 blk(256);
  const float inv_sqrt_kv = 0.032274861218395f;  // 1/sqrt(960)

  // 1) G[b] = emb[b]^T @ emb_all[b]          M=C, N=KV, K=S
  wmma_gemm_f32<<<dim3(KV / BN, C / BM, B), blk, 0, stream>>>(
      emb, emb_all, G, C, KV, S,
      1L, (long)C, (long)S * C, 0L,
      (long)KV, 1L, (long)S * KV, 0L,
      (long)KV, plane, 0L, 1, 1.0f);

  // 2) SA[b,h] = Wq[h] @ G[b]                M=C, N=KV, K=C
  wmma_gemm_f32<<<dim3(KV / BN, C / BM, B * H), blk, 0, stream>>>(
      Wq, G, SA, C, KV, C,
      (long)C, 1L, 0L, (long)C * C,
      (long)KV, 1L, plane, 0L,
      (long)KV, (long)H * plane, plane, H, 1.0f);

  // 3) SB[b,h] = (SA[b,h] @ Wk[h]^T) / sqrt(KV)   M=C, N=KV, K=KV
  wmma_gemm_f32<<<dim3(KV / BN, C / BM, B * H), blk, 0, stream>>>(
      SA, Wk, SB, C, KV, KV,
      (long)KV, 1L, (long)H * plane, plane,
      1L, (long)KV, 0L, (long)KV * KV,
      (long)KV, (long)H * plane, plane, H, inv_sqrt_kv);

  // 4) InstanceNorm stats (two-stage, deterministic)
  inorm_partial<<<dim3(B * H, NCH), blk, 0, stream>>>(SB, part);
  inorm_finalize<<<B * H, 64, 0, stream>>>(part, stats);

  // 5) normalize + softmax over KV, in place on SB
  inorm_softmax<<<B * H * C, blk, 0, stream>>>(SB, stats, C, KV);

  // 6) SA[b,h] = (probs[b,h] @ Wv[h]) * (1/H)     M=C, N=KV, K=KV
  wmma_gemm_f32<<<dim3(KV / BN, C / BM, B * H), blk, 0, stream>>>(
      SB, Wv, SA, C, KV, KV,
      (long)KV, 1L, (long)H * plane, plane,
      (long)KV, 1L, 0L, (long)KV * KV,
      (long)KV, (long)H * plane, plane, H, 0.25f);

  // 7) Pp[b] = sum_h SA[b,h]
  head_sum<<<(unsigned)((B * plane) / 256), blk, 0, stream>>>(SA, Pp, plane, H);

  // 8) Rb[b] = Wo @ Pp[b]                    M=C, N=KV, K=C
  wmma_gemm_f32<<<dim3(KV / BN, C / BM, B), blk, 0, stream>>>(
      Wo, Pp, Rb, C, KV, C,
      (long)C, 1L, 0L, 0L,
      (long)KV, 1L, plane, 0L,
      (long)KV, plane, 0L, 1, 1.0f);

  // 9) out[b] = emb_all[b] @ Rb[b]^T         M=S, N=C, K=KV
  wmma_gemm_f32<<<dim3(C / BN, S / BM, B), blk, 0, stream>>>(
      emb_all, Rb, out, S, C, KV,
      (long)KV, 1L, (long)S * KV, 0L,
      1L, (long)KV, plane, 0L,
      (long)C, (long)S * C, 0L, 1, 1.0f);
}